// SS_21345987461565
// MI455X (gfx1250) — compile-verified
//
#include <hip/hip_runtime.h>
#include <cstdint>

// Problem geometry (from reference): rec_audio is (256, 131072) fp32,
// k = int(0.03 * 131072) = 3932 zeroed per row.
#define N_COLS      131072u
#define N_ROWS      256u
#define K_SEL       3932u
#define F4_PER_ROW  (N_COLS / 4u)        // 32768 float4 per row
#define NBINS       4096
#define T1_THREADS  512
#define CAND_MAX    1024
#define ITERS       8                    // float4 iterations per thread in pass 2
#define NBUF        4                    // async pipeline depth (ring buffers)
#define T2_THREADS  256

// Native clang vector type: accepted by __builtin_nontemporal_store
// (HIP's float4 struct is not).
typedef float f32x4 __attribute__((ext_vector_type(4)));

// Cheap deterministic counter-based PRNG (murmur3-finalizer style).
// ~8 VALU ops/element keeps the elementwise pass memory-bound.
__device__ __forceinline__ uint32_t hash32(uint32_t row, uint32_t col) {
  uint32_t h = row * 0x9E3779B1u + col * 0x85EBCA77u + 42u * 0xC2B2AE3Du;
  h ^= h >> 16;  h *= 0x7FEB352Du;
  h ^= h >> 15;  h *= 0x846CA68Bu;
  h ^= h >> 16;
  return h;
}

// 49-bit strictly-distinct key: (score:32, col:17). Distinctness makes
// "zero iff key >= T(row)" select exactly K_SEL elements per row.
__device__ __forceinline__ unsigned long long make_key(uint32_t h, uint32_t col) {
  return ((unsigned long long)h << 17) | (unsigned long long)col;
}

// ---------------------------------------------------------------------------
// Pass 1: per-row exact top-k radix select. One workgroup per row.
//   Phase A: LDS histogram of top-12 score bits (ds_add atomics).
//   Phase B: descending scan -> cutoff bin b0 and residual rank r inside it.
//   Phase C: gather the ~32 candidates of bin b0, serial-select the r-th
//            largest 49-bit key -> per-row threshold in d_ws.
// ---------------------------------------------------------------------------
__global__ void __launch_bounds__(T1_THREADS)
row_threshold_kernel(unsigned long long* __restrict__ thr) {
  __shared__ unsigned int       hist[NBINS];
  __shared__ unsigned int       partial[T1_THREADS];
  __shared__ unsigned long long cand[CAND_MAX];
  __shared__ unsigned int       s_b0, s_r, s_cnt;

  const unsigned tid = threadIdx.x;
  const unsigned row = blockIdx.x;

  for (unsigned i = tid; i < NBINS; i += T1_THREADS) hist[i] = 0u;
  if (tid == 0) s_cnt = 0u;
  __syncthreads();

  // Phase A: histogram of score[31:20]
  for (unsigned col = tid; col < N_COLS; col += T1_THREADS) {
    const uint32_t h = hash32(row, col);
    atomicAdd(&hist[h >> 20], 1u);
  }
  __syncthreads();

  // Phase B: descending chunk sums (8 bins/thread), then serial crossing scan.
  {
    unsigned s = 0;
    const int hi = NBINS - 1 - (int)tid * 8;
    #pragma unroll
    for (int j = 0; j < 8; ++j) s += hist[hi - j];
    partial[tid] = s;
  }
  __syncthreads();

  if (tid == 0) {
    unsigned cum = 0;
    bool found = false;
    for (int t = 0; t < T1_THREADS && !found; ++t) {
      if (cum + partial[t] >= K_SEL) {
        const int h2 = NBINS - 1 - t * 8;
        for (int j = 0; j < 8; ++j) {
          const unsigned c = hist[h2 - j];
          if (cum + c >= K_SEL) {
            s_b0 = (unsigned)(h2 - j);   // cutoff bin
            s_r  = K_SEL - cum;          // rank needed inside cutoff bin (>=1)
            found = true;
            break;
          }
          cum += c;
        }
      } else {
        cum += partial[t];
      }
    }
  }
  __syncthreads();

  // Phase C: gather candidates of the cutoff bin (expected ~N_COLS/NBINS = 32).
  const unsigned b0 = s_b0;
  for (unsigned col = tid; col < N_COLS; col += T1_THREADS) {
    const uint32_t h = hash32(row, col);
    if ((h >> 20) == b0) {
      const unsigned p = atomicAdd(&s_cnt, 1u);
      if (p < CAND_MAX) cand[p] = make_key(h, col);
    }
  }
  __syncthreads();

  if (tid == 0) {
    const unsigned c = s_cnt < CAND_MAX ? s_cnt : CAND_MAX;
    const unsigned r = s_r;                  // r-th largest among candidates
    unsigned long long T = 0ull;
    for (unsigned i = 0; i < r; ++i) {       // tiny: O(r*c), r,c ~ 32
      unsigned long long mx = 0ull; unsigned mj = 0u;
      for (unsigned j = 0; j < c; ++j)
        if (cand[j] > mx) { mx = cand[j]; mj = j; }
      T = mx;
      cand[mj] = 0ull;
    }
    thr[row] = T;
  }
}

// ---------------------------------------------------------------------------
// Pass 2: bandwidth-bound masked multiply streamed through LDS with the
// CDNA5 async global->LDS path (ASYNCcnt), 4-deep ring pipeline (3 loads
// in flight -> 12 KB outstanding per workgroup to hide HBM latency).
// Each lane stages only its own 16B slot, so only s_wait_asynccnt is needed
// (no workgroup barrier between stage and consume; ASYNC loads complete
// in order, so "asynccnt <= NBUF-1" guarantees the oldest load finished).
// Output uses non-temporal stores: in+out = 268MB > 192MB L2 with zero
// reuse, so keep the store stream from displacing the read stream.
// ---------------------------------------------------------------------------
__global__ void __launch_bounds__(T2_THREADS)
mask_mul_kernel(const float4* __restrict__ in,
                float* __restrict__ out,
                const unsigned long long* __restrict__ thr) {
  __shared__ float4 stage[NBUF][T2_THREADS];

  const unsigned tid = threadIdx.x;
  const unsigned g0  = blockIdx.x * (T2_THREADS * ITERS);   // first float4 of block
  const unsigned row = g0 / F4_PER_ROW;                     // block lies in one row
  const unsigned long long T = thr[row];

  // LDS byte offsets of this lane's staging slots (generic ptr low 32 bits
  // are the wave-relative LDS byte address per the flat-aperture mapping).
  uint32_t lds_off[NBUF];
  #pragma unroll
  for (int b = 0; b < NBUF; ++b)
    lds_off[b] = (uint32_t)(uintptr_t)&stage[b][tid];

  // Prime the pipeline: NBUF-1 async loads in flight.
  #pragma unroll
  for (int it = 0; it < NBUF - 1; ++it) {
    const float4* gp = in + (g0 + (unsigned)it * T2_THREADS + tid);
    asm volatile("global_load_async_to_lds_b128 %0, %1, off"
                 :: "v"(lds_off[it]), "v"(gp) : "memory");
  }

  #pragma unroll
  for (int it = 0; it < ITERS; ++it) {
    if (it + (NBUF - 1) < ITERS) {
      const float4* gn = in + (g0 + (unsigned)(it + NBUF - 1) * T2_THREADS + tid);
      asm volatile("global_load_async_to_lds_b128 %0, %1, off"
                   :: "v"(lds_off[(it + NBUF - 1) % NBUF]), "v"(gn) : "memory");
      // up to NBUF outstanding; wait until the oldest (iteration it) is done
      asm volatile("s_wait_asynccnt %0" :: "i"(NBUF - 1) : "memory");
    } else {
      asm volatile("s_wait_asynccnt 0" ::: "memory");
    }

    const float4 v = stage[it % NBUF][tid];                 // ds_load_b128

    const unsigned g   = g0 + (unsigned)it * T2_THREADS + tid;
    const unsigned col = (g % F4_PER_ROW) * 4u;

    f32x4 r;
    r.x = (make_key(hash32(row, col + 0u), col + 0u) >= T) ? 0.0f : v.x;
    r.y = (make_key(hash32(row, col + 1u), col + 1u) >= T) ? 0.0f : v.y;
    r.z = (make_key(hash32(row, col + 2u), col + 2u) >= T) ? 0.0f : v.z;
    r.w = (make_key(hash32(row, col + 3u), col + 3u) >= T) ? 0.0f : v.w;

    // global_store_b128 with non-temporal hint
    __builtin_nontemporal_store(r, (f32x4*)(out + (size_t)g * 4u));
  }
}

extern "C" void kernel_launch(void* const* d_in, const int* in_sizes, int n_in,
                              void* d_out, int out_size, void* d_ws, size_t ws_size,
                              hipStream_t stream) {
  const float* in  = (const float*)d_in[0];
  float*       out = (float*)d_out;
  unsigned long long* thr = (unsigned long long*)d_ws;  // 256 * 8B = 2 KB scratch

  // Pass 1: one workgroup per row -> exact per-row top-k threshold.
  row_threshold_kernel<<<N_ROWS, T1_THREADS, 0, stream>>>(thr);

  // Pass 2: 256*131072/4 float4 elements, ITERS*T2_THREADS per block.
  const unsigned total_f4 = (N_ROWS * N_COLS) / 4u;                 // 8,388,608
  const unsigned blocks   = total_f4 / (T2_THREADS * ITERS);        // 4096
  mask_mul_kernel<<<blocks, T2_THREADS, 0, stream>>>(
      (const float4*)in, out, thr);
}